// HeagNetConv_43439299232301
// MI455X (gfx1250) — compile-verified
//
#include <hip/hip_runtime.h>

typedef float v2f __attribute__((ext_vector_type(2)));
typedef float v8f __attribute__((ext_vector_type(8)));

constexpr int NUn = 200000;
constexpr int NVn = 100000;
constexpr int NEn = 1000000;

// ---------- order-preserving float <-> uint encoding for atomic max ----------
// enc: monotone map float order -> unsigned order; enc(x) > 0 for all finite x,
// so memset(0) is a valid identity for max.
__device__ __forceinline__ unsigned encf(float f) {
  unsigned u = __float_as_uint(f);
  return (u & 0x80000000u) ? ~u : (u | 0x80000000u);
}
__device__ __forceinline__ float decf(unsigned k) {
  unsigned u = (k & 0x80000000u) ? (k & 0x7FFFFFFFu) : ~k;
  return __uint_as_float(u);
}

// ---------------- scoped atomics (steer toward native HW atomics) ------------
__device__ __forceinline__ void atomAddG(float* p, float v) {
  __hip_atomic_fetch_add(p, v, __ATOMIC_RELAXED, __HIP_MEMORY_SCOPE_AGENT);
}
__device__ __forceinline__ void atomMaxG(unsigned* p, unsigned v) {
  __hip_atomic_fetch_max(p, v, __ATOMIC_RELAXED, __HIP_MEMORY_SCOPE_AGENT);
}
__device__ __forceinline__ void atomAddL(float* p, float v) {
  __hip_atomic_fetch_add(p, v, __ATOMIC_RELAXED, __HIP_MEMORY_SCOPE_WORKGROUP);
}

// ---------------------------- scatter (segment sum/max/count) ----------------
__global__ __launch_bounds__(256) void scatter_kernel(
    const float* __restrict__ xu, const float* __restrict__ xv,
    const float* __restrict__ xe, const int* __restrict__ row,
    const int* __restrict__ col, float* __restrict__ sum_mu,
    unsigned* __restrict__ mk_mu, float* __restrict__ sum_ev,
    unsigned* __restrict__ mk_ev, float* __restrict__ cnt_v,
    float* __restrict__ sum_mv, unsigned* __restrict__ mk_mv,
    float* __restrict__ sum_eu, unsigned* __restrict__ mk_eu,
    float* __restrict__ cnt_u) {
  long long idx = (long long)blockIdx.x * blockDim.x + threadIdx.x;
  if (idx >= (long long)NEn * 32) return;
  int e = (int)(idx >> 5);
  int f = (int)(idx & 31);
  int r = row[e];
  int c = col[e];
  float xuv = xu[(size_t)r * 32 + f];
  float xvv = xv[(size_t)c * 32 + f];
  float xev = xe[(size_t)e * 32 + f];
  size_t oc = (size_t)c * 32 + f;
  size_t orr = (size_t)r * 32 + f;
  unsigned eke = encf(xev);
  atomAddG(&sum_mu[oc], xuv);
  atomMaxG(&mk_mu[oc], encf(xuv));
  atomAddG(&sum_ev[oc], xev);
  atomMaxG(&mk_ev[oc], eke);
  atomAddG(&sum_mv[orr], xvv);
  atomMaxG(&mk_mv[orr], encf(xvv));
  atomAddG(&sum_eu[orr], xev);
  atomMaxG(&mk_eu[orr], eke);
  if (f == 0) {
    atomAddG(&cnt_v[c], 1.0f);
    atomAddG(&cnt_u[r], 1.0f);
  }
}

// ------------------------------ A-matrix loaders -----------------------------
struct NodeAggLoader {  // K = 128: [mean_m | max_m | mean_e | max_e]
  const float* sum_m;
  const unsigned* mk_m;
  const float* sum_e;
  const unsigned* mk_e;
  const float* cnt;
  float invc;
  int nonempty;
  size_t base;
  __device__ void prep(int r) {
    float c = cnt[r];
    invc = 1.0f / fmaxf(c, 1.0f);
    nonempty = c > 0.0f;
    base = (size_t)r * 32;
  }
  __device__ float load(int k) const {
    // branchless: unconditional loads + select (no exec-mask churn)
    if (k < 32) return sum_m[base + k] * invc;
    if (k < 64) {
      float d = decf(mk_m[base + k - 32]);
      return nonempty ? d : 0.0f;
    }
    if (k < 96) return sum_e[base + k - 64] * invc;
    float d = decf(mk_e[base + k - 96]);
    return nonempty ? d : 0.0f;
  }
};

struct EdgeLoader {  // K = 96: [xe | xu[row] | xv[col]]
  const float* xe;
  const float* xu;
  const float* xv;
  const int* row;
  const int* col;
  size_t be, bu, bv;
  __device__ void prep(int e) {
    be = (size_t)e * 32;
    bu = (size_t)row[e] * 32;
    bv = (size_t)col[e] * 32;
  }
  __device__ float load(int k) const {
    if (k < 32) return xe[be + k];
    if (k < 64) return xu[bu + k - 32];
    return xv[bv + k - 64];
  }
};

struct ConcatLoader {  // K = 96: [x (32) | h (64)]
  const float* x;
  const float* h;
  size_t bx, bh;
  __device__ void prep(int r) {
    bx = (size_t)r * 32;
    bh = (size_t)r * 64;
  }
  __device__ float load(int k) const {
    return (k < 32) ? x[bx + k] : h[bh + k - 32];
  }
};

// ------------------- WMMA fp32 GEMM: [nrows x K] @ [K x 64] ------------------
// Block = 128 threads = 4 wave32. A tile (16 x K) is staged once in LDS
// (padded stride => conflict-free ds_load_b64 fragments), W (K x 64) staged
// via b128 copies. Each wave computes one 16x16 tile with
// V_WMMA_F32_16X16X4_F32 (fp32 everywhere: memory-bound workload, keep
// reference precision).
template <int K, class Loader>
__global__ __launch_bounds__(128) void wmma_gemm_kernel(
    Loader ld, const float* __restrict__ W, const float* __restrict__ bias,
    float* __restrict__ out, int nrows, float* __restrict__ colsum,
    float* __restrict__ colsumsq, int accumBN) {
  constexpr int AS = K + 2;  // A stride: odd in 8B units -> no bank conflicts
  __shared__ float Wl[K * 64];
  __shared__ float Al[16 * AS];
  __shared__ float bl[64];
  __shared__ float lsum[64];
  __shared__ float lsq[64];
  const int tid = threadIdx.x;
  const int r0 = blockIdx.x * 16;
  if (r0 >= nrows) return;  // uniform per block
  // stage W (vectorized) + bias + stat accumulators
  {
    const float4* W4 = (const float4*)W;
    float4* Wl4 = (float4*)Wl;
    for (int i = tid; i < K * 16; i += 128) Wl4[i] = W4[i];
  }
  if (tid < 64) {
    bl[tid] = bias[tid];
    lsum[tid] = 0.0f;
    lsq[tid] = 0.0f;
  }
  // stage A tile: 8 threads per row, coalesced-ish per row chunk
  {
    const int rl = tid >> 3;
    const int l8 = tid & 7;
    Loader l = ld;
    l.prep(r0 + rl);
    for (int k = l8; k < K; k += 8) Al[rl * AS + k] = l.load(k);
  }
  __syncthreads();

  const int wave = tid >> 5;
  const int lane = tid & 31;
  const int half = lane >> 4;  // 0: lanes 0-15, 1: lanes 16-31
  const int mrow = lane & 15;
  const int n0 = wave * 16;
  const int cidx = n0 + mrow;

  v8f c = {0.f, 0.f, 0.f, 0.f, 0.f, 0.f, 0.f, 0.f};
#pragma unroll
  for (int k = 0; k < K; k += 4) {
    const int ka = k + 2 * half;
    // A 16x4 frag: lane m, VGPR0 = K(2*half), VGPR1 = K(2*half+1)
    v2f a = *(const v2f*)&Al[mrow * AS + ka];
    // B 4x16 frag: VGPR0 = row K(2*half), VGPR1 = row K(2*half+1)
    v2f b;
    b.x = Wl[ka * 64 + cidx];
    b.y = Wl[(ka + 1) * 64 + cidx];
    c = __builtin_amdgcn_wmma_f32_16x16x4_f32(false, a, false, b, (short)0, c,
                                              false, false);
  }
  const float bb = bl[cidx];
  float s = 0.0f, q = 0.0f;
#pragma unroll
  for (int i = 0; i < 8; ++i) {
    const float y = c[i] + bb;
    const int m = r0 + i + 8 * half;  // C/D layout: row = vgpr + 8*half
    out[(size_t)m * 64 + cidx] = y;
    s += y;
    q += y * y;
  }
  if (accumBN) {
    atomAddL(&lsum[cidx], s);
    atomAddL(&lsq[cidx], q);
    __syncthreads();
    if (tid < 64) {
      atomAddG(&colsum[tid], lsum[tid]);
      atomAddG(&colsumsq[tid], lsq[tid]);
    }
  }
}

// ------------------------- BN (batch stats) + PReLU, in place ----------------
__global__ __launch_bounds__(256) void bn_prelu_kernel(
    float* __restrict__ data, long long nelem, const float* __restrict__ colsum,
    const float* __restrict__ colsumsq, const float* __restrict__ g,
    const float* __restrict__ beta, const float* __restrict__ alpha,
    float invN) {
  long long i = (long long)blockIdx.x * blockDim.x + threadIdx.x;
  if (i >= nelem) return;
  int c = (int)(i & 63);
  float m = colsum[c] * invN;
  float v = colsumsq[c] * invN - m * m;  // biased variance (matches jnp.var)
  float sc = g[c] * rsqrtf(v + 1e-5f);
  float sh = beta[c] - m * sc;
  float y = data[i] * sc + sh;
  data[i] = (y >= 0.0f) ? y : alpha[0] * y;
}

// --------------------------------- launcher ----------------------------------
extern "C" void kernel_launch(void* const* d_in, const int* in_sizes, int n_in,
                              void* d_out, int out_size, void* d_ws,
                              size_t ws_size, hipStream_t stream) {
  (void)in_sizes; (void)n_in; (void)out_size; (void)ws_size;
  const float* xu = (const float*)d_in[0];
  const float* xv = (const float*)d_in[1];
  const float* xe = (const float*)d_in[2];
  const float* W_v2u = (const float*)d_in[3];
  const float* b_v2u = (const float*)d_in[4];
  const float* g_v2u = (const float*)d_in[5];
  const float* beta_v2u = (const float*)d_in[6];
  const float* a_v2u = (const float*)d_in[7];
  const float* W_u2v = (const float*)d_in[8];
  const float* b_u2v = (const float*)d_in[9];
  const float* g_u2v = (const float*)d_in[10];
  const float* beta_u2v = (const float*)d_in[11];
  const float* a_u2v = (const float*)d_in[12];
  const float* W_e = (const float*)d_in[13];
  const float* b_e = (const float*)d_in[14];
  const float* g_e = (const float*)d_in[15];
  const float* beta_e = (const float*)d_in[16];
  const float* a_e = (const float*)d_in[17];
  const float* Wu = (const float*)d_in[18];
  const float* bu = (const float*)d_in[19];
  const float* Wv = (const float*)d_in[20];
  const float* bv = (const float*)d_in[21];
  const int* row = (const int*)d_in[22];
  const int* col = (const int*)d_in[23];

  // ---- workspace layout (floats) ----
  float* ws = (float*)d_ws;
  size_t p = 0;
  float* sum_mu = ws + p;                p += (size_t)NVn * 32;
  unsigned* mk_mu = (unsigned*)(ws + p); p += (size_t)NVn * 32;
  float* sum_ev = ws + p;                p += (size_t)NVn * 32;
  unsigned* mk_ev = (unsigned*)(ws + p); p += (size_t)NVn * 32;
  float* cnt_v = ws + p;                 p += (size_t)NVn;
  float* sum_mv = ws + p;                p += (size_t)NUn * 32;
  unsigned* mk_mv = (unsigned*)(ws + p); p += (size_t)NUn * 32;
  float* sum_eu = ws + p;                p += (size_t)NUn * 32;
  unsigned* mk_eu = (unsigned*)(ws + p); p += (size_t)NUn * 32;
  float* cnt_u = ws + p;                 p += (size_t)NUn;
  float* h_v = ws + p;                   p += (size_t)NVn * 64;
  float* h_u = ws + p;                   p += (size_t)NUn * 64;
  float* stats = ws + p;                 p += 6 * 64;
  // stats: [csum_v|csq_v|csum_u|csq_u|csum_e|csq_e]

  float* out_u = (float*)d_out;
  float* out_v = out_u + (size_t)NUn * 64;
  float* out_e = out_v + (size_t)NVn * 64;

  // 1) zero accumulators + h buffers + BN stats (memset-0 = identity for all)
  hipMemsetAsync(d_ws, 0, p * sizeof(float), stream);

  // 2) edge scatter (segment sum / max / count, both directions)
  {
    long long total = (long long)NEn * 32;
    int blocks = (int)((total + 255) / 256);
    scatter_kernel<<<blocks, 256, 0, stream>>>(
        xu, xv, xe, row, col, sum_mu, mk_mu, sum_ev, mk_ev, cnt_v, sum_mv,
        mk_mv, sum_eu, mk_eu, cnt_u);
  }

  // 3) node GEMMs (pre-BN) + BN stat accumulation
  {
    NodeAggLoader lv{sum_mu, mk_mu, sum_ev, mk_ev, cnt_v, 0.f, 0, 0};
    wmma_gemm_kernel<128, NodeAggLoader><<<NVn / 16, 128, 0, stream>>>(
        lv, W_u2v, b_u2v, h_v, NVn, stats + 0, stats + 64, 1);
    NodeAggLoader lu{sum_mv, mk_mv, sum_eu, mk_eu, cnt_u, 0.f, 0, 0};
    wmma_gemm_kernel<128, NodeAggLoader><<<NUn / 16, 128, 0, stream>>>(
        lu, W_v2u, b_v2u, h_u, NUn, stats + 128, stats + 192, 1);
  }

  // 4) edge GEMM (pre-BN, straight into d_out) + BN stats
  {
    EdgeLoader le{xe, xu, xv, row, col, 0, 0, 0};
    wmma_gemm_kernel<96, EdgeLoader><<<NEn / 16, 128, 0, stream>>>(
        le, W_e, b_e, out_e, NEn, stats + 256, stats + 320, 1);
  }

  // 5) BN + PReLU in place
  {
    long long nv64 = (long long)NVn * 64;
    bn_prelu_kernel<<<(int)((nv64 + 255) / 256), 256, 0, stream>>>(
        h_v, nv64, stats + 0, stats + 64, g_u2v, beta_u2v, a_u2v,
        1.0f / (float)NVn);
    long long nu64 = (long long)NUn * 64;
    bn_prelu_kernel<<<(int)((nu64 + 255) / 256), 256, 0, stream>>>(
        h_u, nu64, stats + 128, stats + 192, g_v2u, beta_v2u, a_v2u,
        1.0f / (float)NUn);
    long long ne64 = (long long)NEn * 64;
    bn_prelu_kernel<<<(int)((ne64 + 255) / 256), 256, 0, stream>>>(
        out_e, ne64, stats + 256, stats + 320, g_e, beta_e, a_e,
        1.0f / (float)NEn);
  }

  // 6) final node lins: concat(x, h) @ W + b  (no BN)
  {
    ConcatLoader cu{xu, h_u, 0, 0};
    wmma_gemm_kernel<96, ConcatLoader><<<NUn / 16, 128, 0, stream>>>(
        cu, Wu, bu, out_u, NUn, stats, stats, 0);
    ConcatLoader cv{xv, h_v, 0, 0};
    wmma_gemm_kernel<96, ConcatLoader><<<NVn / 16, 128, 0, stream>>>(
        cv, Wv, bv, out_v, NVn, stats, stats, 0);
  }
}